// ChebNet_60825326846458
// MI455X (gfx1250) — compile-verified
//
#include <hip/hip_runtime.h>
#include <math.h>

#define BB 8
#define NN 1024
#define CC 64
#define OCC 64
#define KK 5   // K+1

typedef __attribute__((ext_vector_type(2))) float v2f;
typedef __attribute__((ext_vector_type(4))) float v4f;
typedef __attribute__((ext_vector_type(8))) float v8f;

// ---------------------------------------------------------------------------
// Kernel 1: complex projections  src/dst[b,n] = X[b,n,:] (dot) aw vectors
// ---------------------------------------------------------------------------
__global__ void k1_proj(const float* __restrict__ Xr, const float* __restrict__ Xi,
                        const float* __restrict__ awr, const float* __restrict__ awi,
                        float* __restrict__ src_r, float* __restrict__ src_i,
                        float* __restrict__ dst_r, float* __restrict__ dst_i) {
  int id = blockIdx.x * blockDim.x + threadIdx.x; // one (b,n) per thread
  if (id >= BB * NN) return;
  const float* xr = Xr + (size_t)id * CC;
  const float* xi = Xi + (size_t)id * CC;
  float sr = 0.f, si = 0.f, dr = 0.f, di = 0.f;
#pragma unroll 4
  for (int c = 0; c < CC; c += 4) {
    v4f a4 = *(const v4f*)(xr + c);
    v4f b4 = *(const v4f*)(xi + c);
    v4f wsr = *(const v4f*)(awr + c);
    v4f wsi = *(const v4f*)(awi + c);
    v4f wdr = *(const v4f*)(awr + CC + c);
    v4f wdi = *(const v4f*)(awi + CC + c);
#pragma unroll
    for (int j = 0; j < 4; ++j) {
      float a = a4[j], b = b4[j];
      sr += a * wsr[j] - b * wsi[j];
      si += a * wsi[j] + b * wsr[j];
      dr += a * wdr[j] - b * wdi[j];
      di += a * wdi[j] + b * wdr[j];
    }
  }
  src_r[id] = sr; src_i[id] = si; dst_r[id] = dr; dst_i[id] = di;
}

// ---------------------------------------------------------------------------
// Kernel 2: per-column (fixed m) online softmax stats over n of |s[b,n,m]|
// ---------------------------------------------------------------------------
__global__ void __launch_bounds__(256) k2_stats(
    const float* __restrict__ src_r, const float* __restrict__ src_i,
    const float* __restrict__ dst_r, const float* __restrict__ dst_i,
    const float* __restrict__ ab_r_p, const float* __restrict__ ab_i_p,
    const float* __restrict__ pa_r_p, const float* __restrict__ pa_i_p,
    float* __restrict__ colmax, float* __restrict__ colsum) {
  __shared__ float s_sr[NN];
  __shared__ float s_si[NN];
  int b = blockIdx.x >> 2;
  int m0 = (blockIdx.x & 3) * 256;
  for (int i = threadIdx.x; i < NN; i += 256) {
    s_sr[i] = src_r[b * NN + i];
    s_si[i] = src_i[b * NN + i];
  }
  __syncthreads();
  float abr = ab_r_p[0], abi = ab_i_p[0], par = pa_r_p[0], pai = pa_i_p[0];
  int m = m0 + threadIdx.x;
  float dr = dst_r[b * NN + m] + abr;
  float di = dst_i[b * NN + m] + abi;
  float mx = -1e30f, sum = 0.f;
  for (int n = 0; n < NN; ++n) {
    float sr = s_sr[n] + dr; sr = sr >= 0.f ? sr : par * sr;
    float si = s_si[n] + di; si = si >= 0.f ? si : pai * si;
    float mag = sqrtf(sr * sr + si * si);
    if (mag > mx) { sum = sum * __expf(mx - mag) + 1.f; mx = mag; }
    else          { sum += __expf(mag - mx); }
  }
  colmax[b * NN + m] = mx;
  colsum[b * NN + m] = sum;
}

// ---------------------------------------------------------------------------
// Kernel 3: THE memory-bound kernel (335 MB L stream). One wave per (b,n) row;
// each lane owns 4 consecutive m values -> all traffic is b128 loads.
// a[n,m] recomputed once and amortized over all 5 k-planes of Lr & Li.
// ---------------------------------------------------------------------------
__global__ void __launch_bounds__(256) k3_SL(
    const float* __restrict__ Lr, const float* __restrict__ Li,
    const float* __restrict__ src_r, const float* __restrict__ src_i,
    const float* __restrict__ dst_r, const float* __restrict__ dst_i,
    const float* __restrict__ colmax, const float* __restrict__ colsum,
    const float* __restrict__ ab_r_p, const float* __restrict__ ab_i_p,
    const float* __restrict__ pa_r_p, const float* __restrict__ pa_i_p,
    float* __restrict__ SLr, float* __restrict__ SLi) {
  int wave = (blockIdx.x * blockDim.x + threadIdx.x) >> 5; // one wave per row
  int lane = threadIdx.x & 31;
  int b = wave / NN, n = wave % NN;
  float abr = ab_r_p[0], abi = ab_i_p[0], par = pa_r_p[0], pai = pa_i_p[0];
  float ssr = src_r[b * NN + n] + abr;   // uniform across the wave
  float ssi = src_i[b * NN + n] + abi;
  float accR[KK], accI[KK];
#pragma unroll
  for (int k = 0; k < KK; ++k) { accR[k] = 0.f; accI[k] = 0.f; }
  const size_t baseL = (size_t)b * KK * NN * NN + (size_t)n * NN;
  const int sb = b * NN;
  for (int mi = 0; mi < NN / 128; ++mi) {
    int m0 = mi * 128 + lane * 4;                 // 16B-aligned, coalesced
    v4f dr4 = *(const v4f*)(dst_r + sb + m0);
    v4f di4 = *(const v4f*)(dst_i + sb + m0);
    v4f cx4 = *(const v4f*)(colmax + sb + m0);
    v4f cs4 = *(const v4f*)(colsum + sb + m0);
    v4f arv, aiv;
#pragma unroll
    for (int j = 0; j < 4; ++j) {
      float sr = ssr + dr4[j]; sr = sr >= 0.f ? sr : par * sr;
      float si = ssi + di4[j]; si = si >= 0.f ? si : pai * si;
      float mag = sqrtf(sr * sr + si * si);
      float sm = __expf(mag - cx4[j]) / cs4[j];
      float safe = (mag == 0.f) ? 1.f : mag;
      float inv = sm / safe;
      arv[j] = inv * sr;
      aiv[j] = inv * si;
    }
#pragma unroll
    for (int k = 0; k < KK; ++k) {
      size_t idx = baseL + (size_t)k * NN * NN + m0;
      v4f lr4 = *(const v4f*)(Lr + idx);          // b128, streams HBM
      v4f li4 = *(const v4f*)(Li + idx);
      float r = accR[k], i2 = accI[k];
#pragma unroll
      for (int j = 0; j < 4; ++j) {
        r  += lr4[j] * arv[j] - li4[j] * aiv[j];
        i2 += lr4[j] * aiv[j] + li4[j] * arv[j];
      }
      accR[k] = r; accI[k] = i2;
    }
  }
#pragma unroll
  for (int k = 0; k < KK; ++k) {
    float r = accR[k], i = accI[k];
    for (int off = 16; off; off >>= 1) {
      r += __shfl_xor(r, off, 32);
      i += __shfl_xor(i, off, 32);
    }
    if (lane == 0) {
      SLr[(b * KK + k) * NN + n] = r;
      SLi[(b * KK + k) * NN + n] = i;
    }
  }
}

// ---------------------------------------------------------------------------
// Kernel 4: out[b,n,o] = sum_k SLr*(Xr@Wr) -/+ SLi*(Xi@Wi) etc. via fp32 WMMA.
// One wave per 16-row n-tile; diag(SL) folded into the A fragment rows.
// v_wmma_f32_16x16x4_f32: A = 16x4 (2 VGPR: lane<16 holds K0,K1; lane>=16 K2,K3)
//                         B = 4x16 (2 VGPR: reg0 rows K0/K2, reg1 rows K1/K3)
//                         C/D = 16x16 (8 VGPR, row j + 8*(lane>=16), col lane&15)
// ---------------------------------------------------------------------------
__global__ void __launch_bounds__(32) k4_out(
    const float* __restrict__ Xr, const float* __restrict__ Xi,
    const float* __restrict__ Wr, const float* __restrict__ Wi,
    const float* __restrict__ SLr, const float* __restrict__ SLi,
    float* __restrict__ out) {
  int tile = blockIdx.x;          // b*64 + ntile
  int b = tile >> 6;
  int n0 = (tile & 63) << 4;
  int lane = threadIdx.x;
  int row = lane & 15;
  int khalf = (lane >> 4) << 1;   // 0 or 2
  v8f accR[4], accI[4];
#pragma unroll
  for (int i = 0; i < 4; ++i) {
    accR[i] = (v8f){0.f,0.f,0.f,0.f,0.f,0.f,0.f,0.f};
    accI[i] = (v8f){0.f,0.f,0.f,0.f,0.f,0.f,0.f,0.f};
  }
  const int n = n0 + row;
  const size_t xbase = ((size_t)b * NN + n) * CC;
  for (int k = 0; k < KK; ++k) {
    float slr = SLr[(b * KK + k) * NN + n];
    float sli = SLi[(b * KK + k) * NN + n];
    float msli = -sli;
    const float* wr = Wr + (size_t)k * CC * OCC;
    const float* wi = Wi + (size_t)k * CC * OCC;
#pragma unroll 4
    for (int cc = 0; cc < CC / 4; ++cc) {
      int c0 = cc * 4 + khalf;
      v2f xr = *(const v2f*)(Xr + xbase + c0);
      v2f xi = *(const v2f*)(Xi + xbase + c0);
      v2f aRr = { slr  * xr.x, slr  * xr.y };  // -> accR via Wr
      v2f aRi = { msli * xi.x, msli * xi.y };  // -> accR via Wi
      v2f aIr = { sli  * xr.x, sli  * xr.y };  // -> accI via Wr
      v2f aIi = { slr  * xi.x, slr  * xi.y };  // -> accI via Wi
#pragma unroll
      for (int ot = 0; ot < 4; ++ot) {
        int o = ot * 16 + row;
        v2f bwr = { wr[(c0 + 0) * OCC + o], wr[(c0 + 1) * OCC + o] };
        v2f bwi = { wi[(c0 + 0) * OCC + o], wi[(c0 + 1) * OCC + o] };
        accR[ot] = __builtin_amdgcn_wmma_f32_16x16x4_f32(false, aRr, false, bwr, (short)0, accR[ot], false, false);
        accR[ot] = __builtin_amdgcn_wmma_f32_16x16x4_f32(false, aRi, false, bwi, (short)0, accR[ot], false, false);
        accI[ot] = __builtin_amdgcn_wmma_f32_16x16x4_f32(false, aIr, false, bwr, (short)0, accI[ot], false, false);
        accI[ot] = __builtin_amdgcn_wmma_f32_16x16x4_f32(false, aIi, false, bwi, (short)0, accI[ot], false, false);
      }
    }
  }
  float* outR = out;
  float* outI = out + (size_t)BB * NN * OCC;
  int rbase = n0 + ((lane >> 4) * 8);
  int col = lane & 15;
#pragma unroll
  for (int j = 0; j < 8; ++j) {
    int rn = rbase + j;
#pragma unroll
    for (int ot = 0; ot < 4; ++ot) {
      size_t oidx = ((size_t)b * NN + rn) * OCC + ot * 16 + col;
      outR[oidx] = accR[ot][j];
      outI[oidx] = accI[ot][j];
    }
  }
}

// ---------------------------------------------------------------------------
extern "C" void kernel_launch(void* const* d_in, const int* in_sizes, int n_in,
                              void* d_out, int out_size, void* d_ws, size_t ws_size,
                              hipStream_t stream) {
  const float* Xr  = (const float*)d_in[0];
  const float* Xi  = (const float*)d_in[1];
  const float* Lr  = (const float*)d_in[2];
  const float* Li  = (const float*)d_in[3];
  const float* Wr  = (const float*)d_in[4];
  const float* Wi  = (const float*)d_in[5];
  const float* awr = (const float*)d_in[6];
  const float* awi = (const float*)d_in[7];
  const float* abr = (const float*)d_in[8];
  const float* abi = (const float*)d_in[9];
  const float* par = (const float*)d_in[10];
  const float* pai = (const float*)d_in[11];
  float* out = (float*)d_out;

  float* ws    = (float*)d_ws;
  float* src_r = ws;              // 8192
  float* src_i = ws + 8192;       // 8192
  float* dst_r = ws + 16384;      // 8192
  float* dst_i = ws + 24576;      // 8192
  float* cmx   = ws + 32768;      // 8192
  float* csm   = ws + 40960;      // 8192
  float* SLr   = ws + 49152;      // 40960
  float* SLi   = ws + 90112;      // 40960

  k1_proj <<<32, 256, 0, stream>>>(Xr, Xi, awr, awi, src_r, src_i, dst_r, dst_i);
  k2_stats<<<32, 256, 0, stream>>>(src_r, src_i, dst_r, dst_i, abr, abi, par, pai, cmx, csm);
  k3_SL  <<<1024, 256, 0, stream>>>(Lr, Li, src_r, src_i, dst_r, dst_i, cmx, csm,
                                    abr, abi, par, pai, SLr, SLi);
  k4_out  <<<512, 32, 0, stream>>>(Xr, Xi, Wr, Wi, SLr, SLi, out);
}